// GraphConvolution_84937273246259
// MI455X (gfx1250) — compile-verified
//
#include <hip/hip_runtime.h>
#include <hip/hip_bf16.h>

typedef __attribute__((ext_vector_type(2))) float v2f;
typedef __attribute__((ext_vector_type(8))) float v8f;

#define FDIM 128

// ---------------------------------------------------------------------------
// Stage 1: support = X @ W  (fp32 WMMA, 16x16x4)
// Block: 256 threads = 8 waves. Block computes a 16-row x 128-col tile.
// Wave w computes the 16x16 tile at columns [16w, 16w+16).
// ---------------------------------------------------------------------------
__global__ void gcn_gemm_xw(const float* __restrict__ X,
                            const float* __restrict__ W,
                            float* __restrict__ S,
                            int nrows) {
    __shared__ float xs[16 * FDIM];          // 8 KB tile of X

    const int tid  = threadIdx.x;
    const int lane = tid & 31;
    const int wv   = tid >> 5;               // 0..7 -> N tile
    const int half = lane >> 4;              // 0 or 1
    const int l    = lane & 15;
    const int row0 = blockIdx.x * 16;
    const int col0 = wv * 16;

    // Cooperative load of the 16x128 X tile into LDS (float4 = 32 per row).
    {
        const float4* xg = reinterpret_cast<const float4*>(X) +
                           (size_t)row0 * (FDIM / 4);
        float4* xl = reinterpret_cast<float4*>(xs);
        #pragma unroll
        for (int i = tid; i < 16 * (FDIM / 4); i += 256) {
            int r = i >> 5;                  // row within tile
            if (row0 + r < nrows) xl[i] = xg[i];
        }
    }
    __syncthreads();

    // Accumulator: 16x16 f32 C/D tile = 8 VGPRs.
    v8f c = {};

    // A fragment (16x4 f32): lanes 0-15 hold {K=k, K=k+1}, lanes 16-31 hold
    // {K=k+2, K=k+3}, M = lane%16.  B fragment (4x16 f32) mirrors by row.
    #pragma unroll
    for (int k = 0; k < FDIM; k += 4) {
        v2f a, b;
        const float* xr = &xs[l * FDIM + k + 2 * half];
        a.x = xr[0];
        a.y = xr[1];
        const float* wr = &W[(size_t)(k + 2 * half) * FDIM + col0 + l];
        b.x = wr[0];
        b.y = wr[FDIM];
        c = __builtin_amdgcn_wmma_f32_16x16x4_f32(
                /*neg_a=*/false, a, /*neg_b=*/false, b,
                /*c_mod=*/(short)0, c, /*reuse_a=*/false, /*reuse_b=*/false);
    }

    // D layout: VGPR r -> row (r + 8*half), col = lane%16 within the tile.
    #pragma unroll
    for (int r = 0; r < 8; ++r) {
        int row = row0 + r + 8 * half;
        if (row < nrows) S[(size_t)row * FDIM + col0 + l] = c[r];
    }
}

// ---------------------------------------------------------------------------
// Stage 2: zero the output (harness poisons d_out; we must re-zero per call).
// ---------------------------------------------------------------------------
__global__ void gcn_zero_out(float4* __restrict__ out, int n4) {
    int i = blockIdx.x * blockDim.x + threadIdx.x;
    if (i < n4) out[i] = make_float4(0.f, 0.f, 0.f, 0.f);
}

// ---------------------------------------------------------------------------
// Stage 3: out[dst] += edge_weight * support[src]   (wave32 per edge)
// Each lane handles 4 consecutive features: coalesced 512B row per edge.
// Atomics land in L2 (51.2 MB output << 192 MB L2).
// ---------------------------------------------------------------------------
__global__ void gcn_edge_scatter(const int* __restrict__ src,
                                 const int* __restrict__ dst,
                                 const float* __restrict__ ew,
                                 const float* __restrict__ sup,
                                 float* __restrict__ out,
                                 int nedges) {
    int e    = (blockIdx.x * blockDim.x + threadIdx.x) >> 5;
    int lane = threadIdx.x & 31;
    if (e >= nedges) return;

    int   s = src[e];
    int   d = dst[e];
    float w = ew[e];

    float4 v = reinterpret_cast<const float4*>(sup + (size_t)s * FDIM)[lane];
    float* op = out + (size_t)d * FDIM + lane * 4;
    atomicAdd(op + 0, v.x * w);
    atomicAdd(op + 1, v.y * w);
    atomicAdd(op + 2, v.z * w);
    atomicAdd(op + 3, v.w * w);
}

// ---------------------------------------------------------------------------
extern "C" void kernel_launch(void* const* d_in, const int* in_sizes, int n_in,
                              void* d_out, int out_size, void* d_ws, size_t ws_size,
                              hipStream_t stream) {
    const int*   edge_index = (const int*)d_in[0];   // (2, E) flat
    const float* edge_w     = (const float*)d_in[1]; // (E,)
    const float* X          = (const float*)d_in[2]; // (N, 128)
    const float* W          = (const float*)d_in[3]; // (128, 128)
    float*       out        = (float*)d_out;         // (N, 128)
    float*       support    = (float*)d_ws;          // scratch: N*128 floats

    const int E = in_sizes[0] / 2;
    const int N = in_sizes[2] / FDIM;

    // 1) support = X @ W   (fp32 WMMA)
    const int row_tiles = (N + 15) / 16;
    gcn_gemm_xw<<<row_tiles, 256, 0, stream>>>(X, W, support, N);

    // 2) out = 0
    const int n4 = out_size / 4;
    gcn_zero_out<<<(n4 + 255) / 256, 256, 0, stream>>>(
        reinterpret_cast<float4*>(out), n4);

    // 3) scatter-add over edges (8 edges per 256-thread block)
    const int blocks = (E + 7) / 8;
    gcn_edge_scatter<<<blocks, 256, 0, stream>>>(
        edge_index, edge_index + E, edge_w, support, out, E);
}